// SpatialGCN_65377992179783
// MI455X (gfx1250) — compile-verified
//
#include <hip/hip_runtime.h>
#include <hip/hip_bf16.h>

typedef __attribute__((ext_vector_type(2))) float v2f;
typedef __attribute__((ext_vector_type(8))) float v8f;

#define NNODES 50000
#define NEDGES 800000

// ---------------------------------------------------------------------------
// Utility: zero a float buffer (grid-stride)
// ---------------------------------------------------------------------------
__global__ void zero_f32(float* __restrict__ p, long n) {
  long i = (long)blockIdx.x * blockDim.x + threadIdx.x;
  long stride = (long)gridDim.x * blockDim.x;
  for (; i < n; i += stride) p[i] = 0.0f;
}

// ---------------------------------------------------------------------------
// Degree accumulation: deg[dst[e]] += w[e]   (self-loop +1 added later)
// ---------------------------------------------------------------------------
__global__ void deg_kernel(const int* __restrict__ dst, const float* __restrict__ w,
                           float* __restrict__ deg, int ne) {
  int e = blockIdx.x * blockDim.x + threadIdx.x;
  if (e < ne) atomicAdd(&deg[dst[e]], w[e]);
}

// dinv[i] = (deg[i] + 1)^-0.5  (in place over deg), selfnorm[i] = dinv[i]^2
__global__ void dinv_kernel(float* __restrict__ deg, float* __restrict__ selfn, int n) {
  int i = blockIdx.x * blockDim.x + threadIdx.x;
  if (i >= n) return;
  float d = deg[i] + 1.0f;           // self loop weight 1 -> deg always > 0
  float di = rsqrtf(d);
  deg[i] = di;
  selfn[i] = di * di;
}

// norm[e] = dinv[src]*w*dinv[dst]
__global__ void norm_kernel(const int* __restrict__ src, const int* __restrict__ dst,
                            const float* __restrict__ w, const float* __restrict__ dinv,
                            float* __restrict__ norm, int ne) {
  int e = blockIdx.x * blockDim.x + threadIdx.x;
  if (e >= ne) return;
  norm[e] = dinv[src[e]] * w[e] * dinv[dst[e]];
}

// ---------------------------------------------------------------------------
// GEMM: H[nrows x mout] = X[nrows x K] @ W[K x mout] using fp32 WMMA 16x16x4.
// One wave computes one 16x16 output tile. All guards are wave-uniform so
// EXEC is all-ones around the WMMA as required by the ISA.
//
// A (16x4 f32): lane m (0-15), VGPR0=K(kk), VGPR1=K(kk+1), kk = (lane>=16)?2:0
// B (4x16 f32): lane n (0-15), VGPR0=K(kk), VGPR1=K(kk+1)
// C/D (16x16 f32): 8 VGPRs; VGPR r = row r (lanes 0-15) / row r+8 (lanes 16-31)
// ---------------------------------------------------------------------------
__global__ void gemm_wmma_f32(const float* __restrict__ X, const float* __restrict__ W,
                              float* __restrict__ H, int K, int mout, int n_tiles) {
  int wave = (blockIdx.x * blockDim.x + threadIdx.x) >> 5;
  int lane = threadIdx.x & 31;
  if (wave >= n_tiles) return;                 // wave-uniform
  int tiles_n = mout >> 4;
  int row_t = wave / tiles_n;
  int col_t = wave - row_t * tiles_n;

  int m  = lane & 15;
  int kk = (lane >> 4) << 1;                   // 0 for lanes 0-15, 2 for lanes 16-31
  const float* xrow = X + (long)(row_t * 16 + m) * K;
  int ncol = col_t * 16 + m;                   // B column index for this lane

  v8f c = {};
  for (int k = 0; k < K; k += 4) {
    v2f a, b;
    a.x = xrow[k + kk];
    a.y = xrow[k + kk + 1];
    b.x = W[(long)(k + kk) * mout + ncol];
    b.y = W[(long)(k + kk + 1) * mout + ncol];
    c = __builtin_amdgcn_wmma_f32_16x16x4_f32(false, a, false, b, (short)0, c,
                                              false, false);
  }

  int n_out = col_t * 16 + (lane & 15);
  int m_base = row_t * 16 + ((lane >> 4) << 3);
#pragma unroll
  for (int r = 0; r < 8; ++r)
    H[(long)(m_base + r) * mout + n_out] = c[r];
}

// ---------------------------------------------------------------------------
// Edge scatter: agg[dst] += norm[e] * H[src].  One wave per edge; each lane
// handles mout/32 contiguous features (coalesced gather, f32 atomics in L2).
// ---------------------------------------------------------------------------
__global__ void scatter_kernel(const float* __restrict__ H, const float* __restrict__ norm,
                               const int* __restrict__ src, const int* __restrict__ dst,
                               float* __restrict__ agg, int ne, int mout) {
  int tid = blockIdx.x * blockDim.x + threadIdx.x;
  int e = tid >> 5;
  if (e >= ne) return;
  int lane = tid & 31;
  int C = mout >> 5;                           // 4 (mout=128) or 2 (mout=64)
  float w = norm[e];
  const float* hp = H + (long)src[e] * mout + lane * C;
  float* op = agg + (long)dst[e] * mout + lane * C;
#pragma unroll 4
  for (int i = 0; i < C; ++i) atomicAdd(op + i, w * hp[i]);
}

// ---------------------------------------------------------------------------
// Fused self-loop + bias + LayerNorm (+ReLU). One wave per node.
// ---------------------------------------------------------------------------
__global__ void agg_ln_kernel(const float* __restrict__ agg, const float* __restrict__ H,
                              const float* __restrict__ selfn, const float* __restrict__ bias,
                              const float* __restrict__ g, const float* __restrict__ t,
                              float* __restrict__ out, int n, int mout, int do_relu) {
  int wave = (blockIdx.x * blockDim.x + threadIdx.x) >> 5;
  int lane = threadIdx.x & 31;
  if (wave >= n) return;                        // wave-uniform
  int C = mout >> 5;
  float sn = selfn[wave];
  const float* ap = agg + (long)wave * mout;
  const float* hp = H + (long)wave * mout;
  float v[4];
  float s = 0.0f;
  for (int i = 0; i < C; ++i) {
    int f = lane * C + i;
    v[i] = ap[f] + sn * hp[f] + bias[f];
    s += v[i];
  }
#pragma unroll
  for (int msk = 16; msk > 0; msk >>= 1) s += __shfl_xor(s, msk, 32);
  float mean = s / (float)mout;
  float var = 0.0f;
  for (int i = 0; i < C; ++i) { float d = v[i] - mean; var += d * d; }
#pragma unroll
  for (int msk = 16; msk > 0; msk >>= 1) var += __shfl_xor(var, msk, 32);
  float inv = rsqrtf(var / (float)mout + 1e-5f);
  float* op = out + (long)wave * mout;
  for (int i = 0; i < C; ++i) {
    int f = lane * C + i;
    float y = (v[i] - mean) * inv * g[f] + t[f];
    if (do_relu) y = fmaxf(y, 0.0f);
    op[f] = y;
  }
}

// ---------------------------------------------------------------------------
// Host orchestration
// ---------------------------------------------------------------------------
extern "C" void kernel_launch(void* const* d_in, const int* in_sizes, int n_in,
                              void* d_out, int out_size, void* d_ws, size_t ws_size,
                              hipStream_t stream) {
  const float* x  = (const float*)d_in[0];
  const int* ei   = (const int*)d_in[1];
  const int* src  = ei;
  const int* dst  = ei + NEDGES;
  const float* ew = (const float*)d_in[2];
  const float* W1 = (const float*)d_in[3];  const float* b1 = (const float*)d_in[4];
  const float* g1 = (const float*)d_in[5];  const float* t1 = (const float*)d_in[6];
  const float* W2 = (const float*)d_in[7];  const float* b2 = (const float*)d_in[8];
  const float* g2 = (const float*)d_in[9];  const float* t2 = (const float*)d_in[10];
  const float* W3 = (const float*)d_in[11]; const float* b3 = (const float*)d_in[12];
  const float* g3 = (const float*)d_in[13]; const float* t3 = (const float*)d_in[14];

  const int N = NNODES, E = NEDGES;
  float* ws    = (float*)d_ws;
  float* dinv  = ws;                       // N   (holds deg, then dinv)
  float* selfn = dinv + N;                 // N
  float* norm  = selfn + N;                // E
  float* hbuf  = norm + E;                 // N*128
  float* agg   = hbuf + (long)N * 128;     // N*128
  float* xbuf  = agg + (long)N * 128;      // N*128
  float* outp  = (float*)d_out;            // N*64

  // --- graph normalization (once, reused by all 3 layers) ---
  zero_f32<<<1024, 256, 0, stream>>>(dinv, N);
  deg_kernel<<<(E + 255) / 256, 256, 0, stream>>>(dst, ew, dinv, E);
  dinv_kernel<<<(N + 255) / 256, 256, 0, stream>>>(dinv, selfn, N);
  norm_kernel<<<(E + 255) / 256, 256, 0, stream>>>(src, dst, ew, dinv, norm, E);

  const int row_tiles = N / 16;            // 3125 exactly
  const int scat_blocks = (E * 32 + 255) / 256;
  const int ln_blocks = (N * 32 + 255) / 256;

  // --- layer 1: 128 -> 128, relu ---
  {
    int tiles = row_tiles * (128 / 16);
    gemm_wmma_f32<<<(tiles + 7) / 8, 256, 0, stream>>>(x, W1, hbuf, 128, 128, tiles);
    zero_f32<<<2048, 256, 0, stream>>>(agg, (long)N * 128);
    scatter_kernel<<<scat_blocks, 256, 0, stream>>>(hbuf, norm, src, dst, agg, E, 128);
    agg_ln_kernel<<<ln_blocks, 256, 0, stream>>>(agg, hbuf, selfn, b1, g1, t1, xbuf, N, 128, 1);
  }
  // --- layer 2: 128 -> 128, relu ---
  {
    int tiles = row_tiles * (128 / 16);
    gemm_wmma_f32<<<(tiles + 7) / 8, 256, 0, stream>>>(xbuf, W2, hbuf, 128, 128, tiles);
    zero_f32<<<2048, 256, 0, stream>>>(agg, (long)N * 128);
    scatter_kernel<<<scat_blocks, 256, 0, stream>>>(hbuf, norm, src, dst, agg, E, 128);
    agg_ln_kernel<<<ln_blocks, 256, 0, stream>>>(agg, hbuf, selfn, b2, g2, t2, xbuf, N, 128, 1);
  }
  // --- layer 3: 128 -> 64, no relu, straight to d_out ---
  {
    int tiles = row_tiles * (64 / 16);
    gemm_wmma_f32<<<(tiles + 7) / 8, 256, 0, stream>>>(xbuf, W3, hbuf, 128, 64, tiles);
    zero_f32<<<2048, 256, 0, stream>>>(agg, (long)N * 64);
    scatter_kernel<<<scat_blocks, 256, 0, stream>>>(hbuf, norm, src, dst, agg, E, 64);
    agg_ln_kernel<<<ln_blocks, 256, 0, stream>>>(agg, hbuf, selfn, b3, g3, t3, outp, N, 64, 0);
  }
}